// SeqPairInfluenceNN_74929999446226
// MI455X (gfx1250) — compile-verified
//
#include <hip/hip_runtime.h>
#include <hip/hip_bf16.h>

// GRU (P=8192, T=512, I=16, H=32) + 32->3 sigmoid head on gfx1250.
// One wave32 owns a 16-row P-tile and iterates the recurrence over T.
// All matmuls use V_WMMA_F32_16X16X4_F32 (exact f32 on the matrix pipe).
// Biases are preloaded as D-layout C tiles and consumed as the C operand of
// the first WMMA of each accumulation chain (no per-step splat movs).

typedef __attribute__((ext_vector_type(2))) float v2f;   // A/B operand: 16x4 / 4x16 f32
typedef __attribute__((ext_vector_type(8))) float v8f;   // C/D operand: 16x16 f32

#define P_SEQ 8192
#define T_LEN 512
#define I_DIM 16
#define H_DIM 32
#define WAVES 4   // waves per workgroup

static __device__ __forceinline__ v8f splat8(float s) {
  v8f v;
#pragma unroll
  for (int i = 0; i < 8; ++i) v[i] = s;
  return v;
}

// sigmoid(x) = rcp(1 + exp(-x))            -> v_exp_f32 + v_add + v_rcp_f32
static __device__ __forceinline__ float fast_sigmoid(float x) {
  return __builtin_amdgcn_rcpf(1.0f + __expf(-x));
}
// tanh(x) = 1 - 2*rcp(exp(2x) + 1)         -> v_add + v_exp + v_add + v_rcp + v_fma
static __device__ __forceinline__ float fast_tanh(float x) {
  float e = __expf(x + x);
  return fmaf(-2.0f, __builtin_amdgcn_rcpf(e + 1.0f), 1.0f);
}

__global__ __launch_bounds__(WAVES * 32, 1)
void gru_head_kernel(const float* __restrict__ x,
                     const float* __restrict__ w_ih,
                     const float* __restrict__ w_hh,
                     const float* __restrict__ b_ih,
                     const float* __restrict__ b_hh,
                     const float* __restrict__ w_head,
                     const float* __restrict__ b_head,
                     float* __restrict__ out) {
  // per-wave h bounce buffer: 16 rows x 32 cols f32 (D-layout -> A-layout restripe)
  __shared__ float lds_h[WAVES][16][H_DIM];

  const int tid  = threadIdx.x;
  const int wave = tid >> 5;
  const int lane = tid & 31;
  const int half = lane >> 4;   // 0: lanes 0-15, 1: lanes 16-31
  const int l15  = lane & 15;

  const int p0 = (blockIdx.x * WAVES + wave) * 16;   // first P row of this wave's tile
  const size_t OUT2 = (size_t)P_SEQ * T_LEN * 3;     // offset of logits in d_out

  // ------------------------------------------------------------------
  // Loop-invariant B operands (kept in VGPRs for all 512 steps).
  // B layout for V_WMMA_F32_16X16X4_F32 (K=4 x N=16):
  //   reg.x = B[k][n], reg.y = B[k+1][n]  with n = lane&15, k = 4*chunk + 2*(lane>>4)
  // We need B = W^T, i.e. B[k][n] = W[n][k]  (row-major W, consecutive k -> v2f load).
  // ------------------------------------------------------------------
  v2f whhB[6][8];                       // 6 N-tiles (96 outs), 8 K-chunks (H=32)
#pragma unroll
  for (int j = 0; j < 6; ++j)
#pragma unroll
    for (int c = 0; c < 8; ++c) {
      int n = j * 16 + l15;
      int k = 4 * c + 2 * half;
      whhB[j][c] = *(const v2f*)&w_hh[n * H_DIM + k];
    }

  v2f wihB[6][4];                       // 6 N-tiles, 4 K-chunks (I=16)
#pragma unroll
  for (int j = 0; j < 6; ++j)
#pragma unroll
    for (int c = 0; c < 4; ++c) {
      int n = j * 16 + l15;
      int k = 4 * c + 2 * half;
      wihB[j][c] = *(const v2f*)&w_ih[n * I_DIM + k];
    }

  v2f whdB[8];                          // head: only N=0..2 columns live, rest zero
#pragma unroll
  for (int c = 0; c < 8; ++c) {
    v2f v = {0.0f, 0.0f};
    if (l15 < 3) v = *(const v2f*)&w_head[l15 * H_DIM + 4 * c + 2 * half];
    whdB[c] = v;
  }

  // Bias C-tiles in D layout (value depends only on column = lane&15, so the
  // splat across the 8 row-VGPRs is exact). Loaded ONCE, reused every step as
  // the C operand of the first WMMA of each chain.
  v8f biasC_rz[4], biasC_xn[2], biasC_hn[2];
#pragma unroll
  for (int j = 0; j < 4; ++j) biasC_rz[j] = splat8(b_ih[j * 16 + l15] + b_hh[j * 16 + l15]);
#pragma unroll
  for (int j = 0; j < 2; ++j) {
    biasC_xn[j] = splat8(b_ih[64 + j * 16 + l15]);  // n-gate: b_ih outside r*()
    biasC_hn[j] = splat8(b_hh[64 + j * 16 + l15]);  // n-gate: b_hh inside r*()
  }
  const v8f biasC_hd = splat8((l15 < 3) ? b_head[l15] : 0.0f);

  // Per-lane x pointer (row = lane&15), advanced 64B per step.
  const float* xptr = x + (size_t)(p0 + l15) * T_LEN * I_DIM + 2 * half;

  // Per-lane output base pointers (valid when lane&15 < 3), advanced 12B/step.
  // Static per-p element offset p*T*3 (<= 43008B, fits signed 24-bit ioffset).
  float* outh = out + (size_t)(p0 + 8 * half) * T_LEN * 3 + l15;
  float* outl = outh + OUT2;

  // State: h as two 16x16 D tiles (8 VGPRs each) and as 8 A-chunks (v2f each).
  v2f hA[8];
#pragma unroll
  for (int c = 0; c < 8; ++c) hA[c] = (v2f){0.0f, 0.0f};
  v8f hD[2];
  hD[0] = splat8(0.0f);
  hD[1] = splat8(0.0f);

  float* myh = &lds_h[wave][0][0];

#pragma unroll 1
  for (int t = 0; t < T_LEN; ++t) {
    // prefetch next timestep's x row (gfx1250 global_prefetch_b8)
    __builtin_prefetch(xptr + I_DIM, 0, 1);

    // x_t A operands: reg.x = A[row][k], reg.y = A[row][k+1],
    // row = lane&15, k = 4*chunk + 2*(lane>>4)
    v2f xA[4];
#pragma unroll
    for (int c = 0; c < 4; ++c)
      xA[c] = *(const v2f*)(xptr + 4 * c);

    // Accumulators: r,z combined (ih+hh+both biases); n split (xn vs hn).
    // First WMMA of each chain consumes the preloaded bias C-tile directly.
    v8f acc_rz[4], acc_xn[2], acc_hn[2];
#pragma unroll
    for (int j = 0; j < 4; ++j) {
      acc_rz[j] = __builtin_amdgcn_wmma_f32_16x16x4_f32(
          false, xA[0], false, wihB[j][0], (short)0, biasC_rz[j], false, false);
#pragma unroll
      for (int c = 1; c < 4; ++c)
        acc_rz[j] = __builtin_amdgcn_wmma_f32_16x16x4_f32(
            false, xA[c], false, wihB[j][c], (short)0, acc_rz[j], false, false);
#pragma unroll
      for (int c = 0; c < 8; ++c)
        acc_rz[j] = __builtin_amdgcn_wmma_f32_16x16x4_f32(
            false, hA[c], false, whhB[j][c], (short)0, acc_rz[j], false, false);
    }
#pragma unroll
    for (int j = 0; j < 2; ++j) {
      acc_xn[j] = __builtin_amdgcn_wmma_f32_16x16x4_f32(
          false, xA[0], false, wihB[4 + j][0], (short)0, biasC_xn[j], false, false);
#pragma unroll
      for (int c = 1; c < 4; ++c)
        acc_xn[j] = __builtin_amdgcn_wmma_f32_16x16x4_f32(
            false, xA[c], false, wihB[4 + j][c], (short)0, acc_xn[j], false, false);
      acc_hn[j] = __builtin_amdgcn_wmma_f32_16x16x4_f32(
          false, hA[0], false, whhB[4 + j][0], (short)0, biasC_hn[j], false, false);
#pragma unroll
      for (int c = 1; c < 8; ++c)
        acc_hn[j] = __builtin_amdgcn_wmma_f32_16x16x4_f32(
            false, hA[c], false, whhB[4 + j][c], (short)0, acc_hn[j], false, false);
    }

    // Elementwise gates in D layout (each lane/VGPR owns one (m,col) element).
#pragma unroll
    for (int nt = 0; nt < 2; ++nt)
#pragma unroll
      for (int p = 0; p < 8; ++p) {
        float r = fast_sigmoid(acc_rz[nt][p]);
        float z = fast_sigmoid(acc_rz[2 + nt][p]);
        float n = fast_tanh(fmaf(r, acc_hn[nt][p], acc_xn[nt][p]));
        hD[nt][p] = fmaf(z, hD[nt][p] - n, n);   // (1-z)*n + z*h
      }

    // Restripe h: D layout -> row-major LDS -> A-chunk layout (same-wave only).
#pragma unroll
    for (int nt = 0; nt < 2; ++nt)
#pragma unroll
      for (int p = 0; p < 8; ++p) {
        int row = p + 8 * half;                  // D: VGPR p holds M=p / M=p+8
        myh[row * H_DIM + nt * 16 + l15] = hD[nt][p];
      }
    asm volatile("s_wait_dscnt 0" ::: "memory");
#pragma unroll
    for (int c = 0; c < 8; ++c)
      hA[c] = *(const v2f*)&myh[l15 * H_DIM + 4 * c + 2 * half];

    // Head: logits = h @ w_head^T + b_head via WMMA (cols 0..2 valid).
    v8f acc_hd = __builtin_amdgcn_wmma_f32_16x16x4_f32(
        false, hA[0], false, whdB[0], (short)0, biasC_hd, false, false);
#pragma unroll
    for (int c = 1; c < 8; ++c)
      acc_hd = __builtin_amdgcn_wmma_f32_16x16x4_f32(
          false, hA[c], false, whdB[c], (short)0, acc_hd, false, false);

    if (l15 < 3) {                               // predicated stores; EXEC restored after
#pragma unroll
      for (int p = 0; p < 8; ++p) {
        float lg = acc_hd[p];
        float s  = fast_sigmoid(lg);
        s = fminf(fmaxf(s, 1e-6f), 1.0f - 1e-6f);
        __builtin_nontemporal_store(s,  outh + (size_t)p * T_LEN * 3);  // h_seq
        __builtin_nontemporal_store(lg, outl + (size_t)p * T_LEN * 3);  // logits
      }
    }
    xptr += I_DIM;   // +64B
    outh += 3;       // +12B
    outl += 3;
  }
}

extern "C" void kernel_launch(void* const* d_in, const int* in_sizes, int n_in,
                              void* d_out, int out_size, void* d_ws, size_t ws_size,
                              hipStream_t stream) {
  (void)in_sizes; (void)n_in; (void)out_size; (void)d_ws; (void)ws_size;
  const float* x      = (const float*)d_in[0];
  const float* w_ih   = (const float*)d_in[1];
  const float* w_hh   = (const float*)d_in[2];
  const float* b_ih   = (const float*)d_in[3];
  const float* b_hh   = (const float*)d_in[4];
  const float* w_head = (const float*)d_in[5];
  const float* b_head = (const float*)d_in[6];
  float* out = (float*)d_out;

  dim3 grid(P_SEQ / (16 * WAVES));   // 128 blocks
  dim3 block(WAVES * 32);            // 4 wave32s per block
  gru_head_kernel<<<grid, block, 0, stream>>>(x, w_ih, w_hh, b_ih, b_hh,
                                              w_head, b_head, out);
}